// AttnDecoderRNN_1614907703733
// MI455X (gfx1250) — compile-verified
//
#include <hip/hip_runtime.h>
#include <hip/hip_bf16.h>
#include <math.h>

#define H 1024
#define S 8192
#define V 5
#define MAX_LEN 23

typedef float v2f __attribute__((ext_vector_type(2)));
typedef float v8f __attribute__((ext_vector_type(8)));

// ---------------------------------------------------------------------------
// init: h <- encoder_hidden, inp <- 0, done <- 0
// ---------------------------------------------------------------------------
__global__ void init_kernel(const float* __restrict__ enc_hidden,
                            float* __restrict__ h, float* __restrict__ inp,
                            float* __restrict__ done) {
    int j = threadIdx.x;             // 1024 threads
    h[j] = enc_hidden[j];
    if (j < V) inp[j] = 0.0f;
    if (j == 0) done[0] = 0.0f;
}

// ---------------------------------------------------------------------------
// kU = keys(SxH) @ Ua(HxH) + bu   -- fp32 WMMA 16x16x4, one 16x64 strip/wave
// A frag (16x4 f32): lane m = lane&15, khalf = (lane>>4)*2 -> {A[m][k],A[m][k+1]}
// B frag (4x16 f32): lane n = lane&15, khalf same          -> {B[k][n],B[k+1][n]}
// C/D (16x16 f32, 8 VGPRs): vgpr r -> row = r + 8*(lane>>4), col = lane&15
// ---------------------------------------------------------------------------
__global__ void kU_wmma_kernel(const float* __restrict__ keys,
                               const float* __restrict__ Ua,
                               const float* __restrict__ bu,
                               float* __restrict__ kU) {
    const int lane   = threadIdx.x & 31;
    const int waveId = blockIdx.x * (blockDim.x >> 5) + (threadIdx.x >> 5);
    const int NT = H / 64;                 // 16 N-strips
    const int tm = waveId / NT;            // 0..S/16-1
    const int tn = waveId % NT;            // 0..15
    const int m0 = tm * 16;
    const int n0 = tn * 64;
    const int mn    = lane & 15;           // M for A, N for B/C
    const int khalf = (lane >> 4) * 2;     // 0 or 2

    v8f c0 = {}, c1 = {}, c2 = {}, c3 = {};
    for (int k0 = 0; k0 < H; k0 += 4) {
        const float* arow = keys + (size_t)(m0 + mn) * H + (k0 + khalf);
        v2f a; a.x = arow[0]; a.y = arow[1];

        const float* b0p = Ua + (size_t)(k0 + khalf) * H + n0 + mn;
        const float* b1p = b0p + H;
        v2f b0; b0.x = b0p[0];  b0.y = b1p[0];
        v2f b1; b1.x = b0p[16]; b1.y = b1p[16];
        v2f b2; b2.x = b0p[32]; b2.y = b1p[32];
        v2f b3; b3.x = b0p[48]; b3.y = b1p[48];

        c0 = __builtin_amdgcn_wmma_f32_16x16x4_f32(false, a, false, b0, (short)0, c0, false, false);
        c1 = __builtin_amdgcn_wmma_f32_16x16x4_f32(false, a, false, b1, (short)0, c1, false, false);
        c2 = __builtin_amdgcn_wmma_f32_16x16x4_f32(false, a, false, b2, (short)0, c2, false, false);
        c3 = __builtin_amdgcn_wmma_f32_16x16x4_f32(false, a, false, b3, (short)0, c3, false, false);
    }

    const int rowBase = m0 + ((lane >> 4) * 8);
    const int col = lane & 15;
    float bu0 = bu[n0 + col], bu1 = bu[n0 + col + 16];
    float bu2 = bu[n0 + col + 32], bu3 = bu[n0 + col + 48];
    for (int r = 0; r < 8; ++r) {
        float* out = kU + (size_t)(rowBase + r) * H + n0 + col;
        out[0]  = c0[r] + bu0;
        out[16] = c1[r] + bu1;
        out[32] = c2[r] + bu2;
        out[48] = c3[r] + bu3;
    }
}

// ---------------------------------------------------------------------------
// qW = h @ Wa + ba ; emb = inp @ Wl + bl   (grid 4 x 256, one thread per col)
// ---------------------------------------------------------------------------
__global__ void vec_kernel(const float* __restrict__ h,  const float* __restrict__ Wa,
                           const float* __restrict__ ba, const float* __restrict__ inp,
                           const float* __restrict__ Wl, const float* __restrict__ bl,
                           float* __restrict__ qW, float* __restrict__ emb) {
    int j = blockIdx.x * blockDim.x + threadIdx.x;  // 0..H-1, coalesced over Wa cols
    float s1 = ba[j];
    for (int i = 0; i < H; ++i) s1 += h[i] * Wa[(size_t)i * H + j];
    qW[j] = s1;
    float s2 = bl[j];
    for (int v = 0; v < V; ++v) s2 += inp[v] * Wl[(size_t)v * H + j];
    emb[j] = s2;
}

// ---------------------------------------------------------------------------
// e[s] = sum_h tanh(qW[h] + kU[s,h]) * Va[h] + bv   (one block per row s)
// ---------------------------------------------------------------------------
__global__ void scores_kernel(const float* __restrict__ kU, const float* __restrict__ qW,
                              const float* __restrict__ Va, const float* __restrict__ bv,
                              float* __restrict__ e) {
    __shared__ float red[256];
    const int s = blockIdx.x;
    const float* row = kU + (size_t)s * H;
    float acc = 0.0f;
    for (int hh = threadIdx.x; hh < H; hh += 256)
        acc += tanhf(qW[hh] + row[hh]) * Va[hh];
    red[threadIdx.x] = acc;
    __syncthreads();
    for (int off = 128; off > 0; off >>= 1) {
        if (threadIdx.x < off) red[threadIdx.x] += red[threadIdx.x + off];
        __syncthreads();
    }
    if (threadIdx.x == 0) e[s] = red[0] + bv[0];
}

// ---------------------------------------------------------------------------
// w = softmax(e) written straight into the attns output slice; zero context
// ---------------------------------------------------------------------------
__global__ void softmax_kernel(const float* __restrict__ e, float* __restrict__ w_out,
                               float* __restrict__ context) {
    __shared__ float red[1024];
    const int t = threadIdx.x;
    float mx = -INFINITY;
    for (int s = t; s < S; s += 1024) mx = fmaxf(mx, e[s]);
    red[t] = mx; __syncthreads();
    for (int off = 512; off > 0; off >>= 1) {
        if (t < off) red[t] = fmaxf(red[t], red[t + off]);
        __syncthreads();
    }
    mx = red[0]; __syncthreads();
    float sum = 0.0f;
    for (int s = t; s < S; s += 1024) sum += expf(e[s] - mx);
    red[t] = sum; __syncthreads();
    for (int off = 512; off > 0; off >>= 1) {
        if (t < off) red[t] += red[t + off];
        __syncthreads();
    }
    float inv = 1.0f / red[0];
    for (int s = t; s < S; s += 1024) w_out[s] = expf(e[s] - mx) * inv;
    context[t] = 0.0f;   // H == blockDim.x == 1024
}

// ---------------------------------------------------------------------------
// context[h] += sum_{s in chunk} w[s]*keys[s,h]   grid = (H/256, S/256)
// ---------------------------------------------------------------------------
__global__ void context_kernel(const float* __restrict__ w, const float* __restrict__ keys,
                               float* __restrict__ context) {
    const int hh = blockIdx.x * 256 + threadIdx.x;
    const int s0 = blockIdx.y * 256;
    float acc = 0.0f;
    for (int s = s0; s < s0 + 256; ++s)
        acc += w[s] * keys[(size_t)s * H + hh];
    atomicAdd(&context[hh], acc);
}

// ---------------------------------------------------------------------------
// gi[r] = W_ih[r,:] . [emb,context] + b_ih ; gh[r] = W_hh[r,:] . h + b_hh
// one block (256 thr) per row r in [0,3H)
// ---------------------------------------------------------------------------
__global__ void gates_kernel(const float* __restrict__ W_ih, const float* __restrict__ b_ih,
                             const float* __restrict__ W_hh, const float* __restrict__ b_hh,
                             const float* __restrict__ emb,  const float* __restrict__ context,
                             const float* __restrict__ h,
                             float* __restrict__ gi, float* __restrict__ gh) {
    __shared__ float red[256];
    const int r = blockIdx.x;
    const int t = threadIdx.x;
    const float* wrow = W_ih + (size_t)r * (2 * H);
    float acc = 0.0f;
    for (int c = t; c < 2 * H; c += 256) {
        float xv = (c < H) ? emb[c] : context[c - H];
        acc += wrow[c] * xv;
    }
    red[t] = acc; __syncthreads();
    for (int off = 128; off > 0; off >>= 1) {
        if (t < off) red[t] += red[t + off];
        __syncthreads();
    }
    if (t == 0) gi[r] = red[0] + b_ih[r];
    __syncthreads();
    const float* hrow = W_hh + (size_t)r * H;
    acc = 0.0f;
    for (int c = t; c < H; c += 256) acc += hrow[c] * h[c];
    red[t] = acc; __syncthreads();
    for (int off = 128; off > 0; off >>= 1) {
        if (t < off) red[t] += red[t + off];
        __syncthreads();
    }
    if (t == 0) gh[r] = red[0] + b_hh[r];
}

// ---------------------------------------------------------------------------
// GRU combine + output head + EOS early-exit state machine (1 block, H thr)
// ---------------------------------------------------------------------------
__global__ void combine_kernel(const float* __restrict__ gi, const float* __restrict__ gh,
                               float* __restrict__ h_ws, float* __restrict__ inp_ws,
                               float* __restrict__ done_ws,
                               const float* __restrict__ W_out, const float* __restrict__ b_out,
                               float* __restrict__ outs_row, float* __restrict__ hfin,
                               int step, int is_last) {
    __shared__ float sh[1024];
    __shared__ float red[1024];
    __shared__ float logits[V];
    const int j = threadIdx.x;
    const float done = done_ws[0];
    const float hold = h_ws[j];
    const float rg = 1.0f / (1.0f + expf(-(gi[j]         + gh[j])));
    const float zg = 1.0f / (1.0f + expf(-(gi[H + j]     + gh[H + j])));
    const float ng = tanhf(gi[2 * H + j] + rg * gh[2 * H + j]);
    const float hnew  = (1.0f - zg) * ng + zg * hold;
    const float hnext = (done != 0.0f) ? hold : hnew;
    sh[j] = hnext;
    h_ws[j] = hnext;
    if (is_last) hfin[j] = hnext;
    __syncthreads();
    for (int o = 0; o < V; ++o) {
        red[j] = sh[j] * W_out[(size_t)j * V + o];
        __syncthreads();
        for (int off = 512; off > 0; off >>= 1) {
            if (j < off) red[j] += red[j + off];
            __syncthreads();
        }
        if (j == 0) logits[o] = red[0] + b_out[o];
        __syncthreads();
    }
    if (j == 0) {
        float mx = logits[0];
        for (int o = 1; o < V; ++o) mx = fmaxf(mx, logits[o]);
        float p[V]; float se = 0.0f;
        for (int o = 0; o < V; ++o) { p[o] = expf(logits[o] - mx); se += p[o]; }
        for (int o = 0; o < V; ++o) p[o] /= se;
        float pmax = p[0];
        for (int o = 1; o < V; ++o) pmax = fmaxf(pmax, p[o]);
        const int trigger = (step >= 5) && (p[V - 1] >= pmax);
        const float done_now = ((done != 0.0f) || trigger) ? 1.0f : 0.0f;
        const float EOS[V] = {0.0f, 0.0f, 0.0f, 0.0f, 1.0f};
        for (int o = 0; o < V; ++o) {
            outs_row[o] = (done_now != 0.0f) ? EOS[o] : p[o];
            if (!(done_now != 0.0f)) inp_ws[o] = p[o];  // else keep previous inp
        }
        done_ws[0] = done_now;
    }
}

// ---------------------------------------------------------------------------
extern "C" void kernel_launch(void* const* d_in, const int* in_sizes, int n_in,
                              void* d_out, int out_size, void* d_ws, size_t ws_size,
                              hipStream_t stream) {
    (void)in_sizes; (void)n_in; (void)out_size; (void)ws_size;
    const float* keys  = (const float*)d_in[0];   // (1,S,H)
    const float* ehid  = (const float*)d_in[1];   // (1,1,H)
    const float* Wa    = (const float*)d_in[2];
    const float* ba    = (const float*)d_in[3];
    const float* Ua    = (const float*)d_in[4];
    const float* bu    = (const float*)d_in[5];
    const float* Va    = (const float*)d_in[6];
    const float* bv    = (const float*)d_in[7];
    const float* Wl    = (const float*)d_in[8];
    const float* bl    = (const float*)d_in[9];
    const float* W_ih  = (const float*)d_in[10];
    const float* b_ih  = (const float*)d_in[11];
    const float* W_hh  = (const float*)d_in[12];
    const float* b_hh  = (const float*)d_in[13];
    const float* W_out = (const float*)d_in[14];
    const float* b_out = (const float*)d_in[15];

    float* out = (float*)d_out;
    float* outs  = out;                         // 23*V
    float* hfin  = out + MAX_LEN * V;           // H
    float* attns = hfin + H;                    // 23*S

    // workspace layout (floats)
    float* ws      = (float*)d_ws;
    float* kU      = ws;                        // S*H
    float* qW      = kU + (size_t)S * H;        // H
    float* emb     = qW + H;                    // H
    float* e       = emb + H;                   // S
    float* context = e + S;                     // H
    float* gi      = context + H;               // 3H
    float* gh      = gi + 3 * H;                // 3H
    float* h_ws    = gh + 3 * H;                // H
    float* inp_ws  = h_ws + H;                  // V (padded use)
    float* done_ws = inp_ws + 8;                // 1

    init_kernel<<<1, H, 0, stream>>>(ehid, h_ws, inp_ws, done_ws);

    // kU GEMM: (S/16)*(H/64) = 8192 waves, 8 waves per 256-thread block
    kU_wmma_kernel<<<(S / 16) * (H / 64) / 8, 256, 0, stream>>>(keys, Ua, bu, kU);

    for (int step = 0; step < MAX_LEN; ++step) {
        float* w_row = attns + (size_t)step * S;
        vec_kernel<<<H / 256, 256, 0, stream>>>(h_ws, Wa, ba, inp_ws, Wl, bl, qW, emb);
        scores_kernel<<<S, 256, 0, stream>>>(kU, qW, Va, bv, e);
        softmax_kernel<<<1, 1024, 0, stream>>>(e, w_row, context);
        context_kernel<<<dim3(H / 256, S / 256), 256, 0, stream>>>(w_row, keys, context);
        gates_kernel<<<3 * H, 256, 0, stream>>>(W_ih, b_ih, W_hh, b_hh, emb, context, h_ws, gi, gh);
        combine_kernel<<<1, H, 0, stream>>>(gi, gh, h_ws, inp_ws, done_ws, W_out, b_out,
                                            outs + (size_t)step * V, hfin,
                                            step, step == MAX_LEN - 1 ? 1 : 0);
    }
}